// Encoder__25580825215780
// MI455X (gfx1250) — compile-verified
//
#include <hip/hip_runtime.h>

#define N_NODES 100000
#define E_EDGES 1600000
#define DIM 128
#define DOUT 64

typedef float v2f __attribute__((ext_vector_type(2)));
typedef float v8f __attribute__((ext_vector_type(8)));

// ---------------------------------------------------------------------------
// Kernel 1: zero aggr, deg = 1 (self loop)
// ---------------------------------------------------------------------------
__global__ void init_kernel(float* __restrict__ aggr, float* __restrict__ deg) {
    int tid = blockIdx.x * blockDim.x + threadIdx.x;
    if (tid < N_NODES * DIM) aggr[tid] = 0.0f;
    if (tid < N_NODES) deg[tid] = 1.0f;
}

// ---------------------------------------------------------------------------
// Kernel 2: aggr[dst] += x[src]  (per edge, 128 floats in float4 chunks)
//           deg[dst]  += 1
// ---------------------------------------------------------------------------
__global__ void scatter_x_kernel(const float* __restrict__ x,
                                 const long long* __restrict__ ei,
                                 float* __restrict__ aggr,
                                 float* __restrict__ deg) {
    int tid = blockIdx.x * blockDim.x + threadIdx.x;
    int e = tid >> 5;                 // 32 chunks of 4 floats per edge
    if (e >= E_EDGES) return;
    int c = (tid & 31) << 2;
    int s = (int)ei[e];
    int d = (int)ei[E_EDGES + e];
    const float4 v = *(const float4*)(x + (size_t)s * DIM + c);
    float* o = aggr + (size_t)d * DIM + c;
    atomicAdd(o + 0, v.x);
    atomicAdd(o + 1, v.y);
    atomicAdd(o + 2, v.z);
    atomicAdd(o + 3, v.w);
    if (c == 0) atomicAdd(deg + d, 1.0f);
}

// ---------------------------------------------------------------------------
// Kernel 3: deg -> rsqrt(deg)   (deg >= 1 always, matches reference)
// ---------------------------------------------------------------------------
__global__ void rsqrt_kernel(float* __restrict__ deg) {
    int i = blockIdx.x * blockDim.x + threadIdx.x;
    if (i < N_NODES) deg[i] = rsqrtf(deg[i]);
}

// ---------------------------------------------------------------------------
// Kernel 4: fused MLP + GCN-head GEMMs, one wave per 16-node tile.
//   A (x+aggr) staged in LDS; each 16x16 output tile accumulated with
//   v_wmma_f32_16x16x4_f32 over K=128 (32 steps).
// VGPR layouts per CDNA5 ISA 7.12.2:
//   A 16x4 : lane L<16 row=L holds K=k,k+1 ; lane L>=16 row=L-16 holds K=k+2,k+3
//   B 4x16 : v0 = row k (lanes0-15) / row k+2 (lanes16-31); v1 = k+1 / k+3
//   C 16x16: vgpr i = (row i, col=lane) for lanes0-15, (row 8+i, col=lane-16)
// ---------------------------------------------------------------------------
__global__ __launch_bounds__(32) void mlp_wmma_kernel(
    const float* __restrict__ x,     const float* __restrict__ aggr,
    const float* __restrict__ W1,    const float* __restrict__ b1,
    const float* __restrict__ gamma, const float* __restrict__ beta,
    const float* __restrict__ rmean, const float* __restrict__ rvar,
    const float* __restrict__ W2,    const float* __restrict__ b2,
    const float* __restrict__ Wmu,   const float* __restrict__ Wls,
    float* __restrict__ xwmu,        float* __restrict__ xwls) {
    __shared__ float A[16][DIM];
    __shared__ float H[16][DIM];

    const int lane = threadIdx.x;
    const int base = blockIdx.x * 16;           // first node of tile (N divisible by 16)
    const int half = lane >> 4;
    const int r    = lane & 15;

    // ---- stage A = x + aggr for 16 nodes (float4) ----
    for (int q = lane; q < 16 * (DIM / 4); q += 32) {
        int row = q >> 5;                        // 32 float4 per row
        int c4  = (q & 31) << 2;
        size_t off = (size_t)(base + row) * DIM + c4;
        float4 xv = *(const float4*)(x + off);
        float4 av = *(const float4*)(aggr + off);
        *(float4*)(&A[row][c4]) =
            make_float4(xv.x + av.x, xv.y + av.y, xv.z + av.z, xv.w + av.w);
    }
    __syncthreads();

    // ---- GEMM1: H = relu(BN(A @ W1 + b1)) ----
    {
        v8f acc[8] = {};
        for (int kk = 0; kk < 32; ++kk) {
            int k = kk * 4 + 2 * half;
            v2f a; a[0] = A[r][k]; a[1] = A[r][k + 1];
#pragma unroll
            for (int nt = 0; nt < 8; ++nt) {
                int n = nt * 16 + r;
                v2f b; b[0] = W1[k * DIM + n]; b[1] = W1[(k + 1) * DIM + n];
                acc[nt] = __builtin_amdgcn_wmma_f32_16x16x4_f32(
                    false, a, false, b, (short)0, acc[nt], false, false);
            }
        }
#pragma unroll
        for (int nt = 0; nt < 8; ++nt) {
            int n = nt * 16 + r;
            float scale = gamma[n] * rsqrtf(rvar[n] + 1e-5f);
            float shift = beta[n] + (b1[n] - rmean[n]) * scale;
#pragma unroll
            for (int i = 0; i < 8; ++i) {
                float v = acc[nt][i] * scale + shift;
                H[i + 8 * half][n] = v > 0.0f ? v : 0.0f;
            }
        }
    }
    __syncthreads();

    // ---- GEMM2: A = relu(H @ W2 + b2)   (relu twice == relu once) ----
    {
        v8f acc[8] = {};
        for (int kk = 0; kk < 32; ++kk) {
            int k = kk * 4 + 2 * half;
            v2f a; a[0] = H[r][k]; a[1] = H[r][k + 1];
#pragma unroll
            for (int nt = 0; nt < 8; ++nt) {
                int n = nt * 16 + r;
                v2f b; b[0] = W2[k * DIM + n]; b[1] = W2[(k + 1) * DIM + n];
                acc[nt] = __builtin_amdgcn_wmma_f32_16x16x4_f32(
                    false, a, false, b, (short)0, acc[nt], false, false);
            }
        }
#pragma unroll
        for (int nt = 0; nt < 8; ++nt) {
            int n = nt * 16 + r;
            float bias = b2[n];
#pragma unroll
            for (int i = 0; i < 8; ++i) {
                float v = acc[nt][i] + bias;
                A[i + 8 * half][n] = v > 0.0f ? v : 0.0f;
            }
        }
    }
    __syncthreads();

    // ---- heads: xwmu = A @ Wmu, xwls = A @ Wls (no bias here) ----
    {
        v8f accm[4] = {};
        v8f accl[4] = {};
        for (int kk = 0; kk < 32; ++kk) {
            int k = kk * 4 + 2 * half;
            v2f a; a[0] = A[r][k]; a[1] = A[r][k + 1];
#pragma unroll
            for (int nt = 0; nt < 4; ++nt) {
                int n = nt * 16 + r;
                v2f bm; bm[0] = Wmu[k * DOUT + n]; bm[1] = Wmu[(k + 1) * DOUT + n];
                accm[nt] = __builtin_amdgcn_wmma_f32_16x16x4_f32(
                    false, a, false, bm, (short)0, accm[nt], false, false);
                v2f bl; bl[0] = Wls[k * DOUT + n]; bl[1] = Wls[(k + 1) * DOUT + n];
                accl[nt] = __builtin_amdgcn_wmma_f32_16x16x4_f32(
                    false, a, false, bl, (short)0, accl[nt], false, false);
            }
        }
#pragma unroll
        for (int nt = 0; nt < 4; ++nt) {
            int n = nt * 16 + r;
#pragma unroll
            for (int i = 0; i < 8; ++i) {
                int node = base + i + 8 * half;
                xwmu[(size_t)node * DOUT + n] = accm[nt][i];
                xwls[(size_t)node * DOUT + n] = accl[nt][i];
            }
        }
    }
}

// ---------------------------------------------------------------------------
// Kernel 5: out = bias + dinv[i]^2 * xw[i]   (self-loop term + bias init)
// ---------------------------------------------------------------------------
__global__ void gcn_init_kernel(const float* __restrict__ dinv,
                                const float* __restrict__ xwmu,
                                const float* __restrict__ xwls,
                                const float* __restrict__ bmu,
                                const float* __restrict__ bls,
                                float* __restrict__ out) {
    int tid = blockIdx.x * blockDim.x + threadIdx.x;
    if (tid >= N_NODES * DOUT) return;
    int i = tid >> 6;
    int j = tid & 63;
    float w = dinv[i];
    w *= w;
    out[tid] = bmu[j] + w * xwmu[tid];
    out[(size_t)N_NODES * DOUT + tid] = bls[j] + w * xwls[tid];
}

// ---------------------------------------------------------------------------
// Kernel 6: out[dst] += dinv[src]*dinv[dst] * xw[src]  for both heads
// ---------------------------------------------------------------------------
__global__ void gcn_scatter_kernel(const long long* __restrict__ ei,
                                   const float* __restrict__ dinv,
                                   const float* __restrict__ xwmu,
                                   const float* __restrict__ xwls,
                                   float* __restrict__ out) {
    int tid = blockIdx.x * blockDim.x + threadIdx.x;
    int e = tid >> 4;                  // 16 chunks of 4 floats per edge
    if (e >= E_EDGES) return;
    int c = (tid & 15) << 2;
    int s = (int)ei[e];
    int d = (int)ei[E_EDGES + e];
    float w = dinv[s] * dinv[d];

    float4 m = *(const float4*)(xwmu + (size_t)s * DOUT + c);
    float* om = out + (size_t)d * DOUT + c;
    atomicAdd(om + 0, w * m.x);
    atomicAdd(om + 1, w * m.y);
    atomicAdd(om + 2, w * m.z);
    atomicAdd(om + 3, w * m.w);

    float4 l = *(const float4*)(xwls + (size_t)s * DOUT + c);
    float* ol = out + (size_t)N_NODES * DOUT + (size_t)d * DOUT + c;
    atomicAdd(ol + 0, w * l.x);
    atomicAdd(ol + 1, w * l.y);
    atomicAdd(ol + 2, w * l.z);
    atomicAdd(ol + 3, w * l.w);
}

// ---------------------------------------------------------------------------
extern "C" void kernel_launch(void* const* d_in, const int* in_sizes, int n_in,
                              void* d_out, int out_size, void* d_ws, size_t ws_size,
                              hipStream_t stream) {
    const float*     x     = (const float*)d_in[0];
    const long long* ei    = (const long long*)d_in[1];
    const float*     W1    = (const float*)d_in[2];
    const float*     b1    = (const float*)d_in[3];
    const float*     gamma = (const float*)d_in[4];
    const float*     beta  = (const float*)d_in[5];
    const float*     rmean = (const float*)d_in[6];
    const float*     rvar  = (const float*)d_in[7];
    const float*     W2    = (const float*)d_in[8];
    const float*     b2    = (const float*)d_in[9];
    const float*     Wmu   = (const float*)d_in[10];
    const float*     bmu   = (const float*)d_in[11];
    const float*     Wls   = (const float*)d_in[12];
    const float*     bls   = (const float*)d_in[13];
    float* out = (float*)d_out;

    // workspace layout (floats): aggr[N*128] | xwmu[N*64] | xwls[N*64] | deg[N]
    float* ws   = (float*)d_ws;
    float* aggr = ws;
    float* xwmu = aggr + (size_t)N_NODES * DIM;
    float* xwls = xwmu + (size_t)N_NODES * DOUT;
    float* deg  = xwls + (size_t)N_NODES * DOUT;

    const int T = 256;

    // 1) init aggr=0, deg=1
    init_kernel<<<(N_NODES * DIM + T - 1) / T, T, 0, stream>>>(aggr, deg);

    // 2) scatter x[src] into aggr[dst], count degrees
    {
        long long items = (long long)E_EDGES * 32;
        int blocks = (int)((items + T - 1) / T);
        scatter_x_kernel<<<blocks, T, 0, stream>>>(x, ei, aggr, deg);
    }

    // 3) dinv = rsqrt(deg)
    rsqrt_kernel<<<(N_NODES + T - 1) / T, T, 0, stream>>>(deg);

    // 4) fused MLP + head GEMMs via f32 WMMA (one wave per 16 nodes)
    mlp_wmma_kernel<<<N_NODES / 16, 32, 0, stream>>>(
        x, aggr, W1, b1, gamma, beta, rmean, rvar, W2, b2, Wmu, Wls, xwmu, xwls);

    // 5) out = bias + self-loop term
    gcn_init_kernel<<<(N_NODES * DOUT + T - 1) / T, T, 0, stream>>>(
        deg, xwmu, xwls, bmu, bls, out);

    // 6) edge scatter of normalized xw into out (both heads)
    {
        long long items = (long long)E_EDGES * 16;
        int blocks = (int)((items + T - 1) / T);
        gcn_scatter_kernel<<<blocks, T, 0, stream>>>(ei, deg, xwmu, xwls, out);
    }
}